// VectorQuantizer_76544907149316
// MI455X (gfx1250) — compile-verified
//
#include <hip/hip_runtime.h>

typedef __attribute__((ext_vector_type(2))) float v2f;
typedef __attribute__((ext_vector_type(8))) float v8f;
typedef __attribute__((ext_vector_type(4))) int   v4i;

#define EMBED_DIM   256
#define NUM_CODES   8192
#define N_ROWS      16384             // 16 * 1024
#define TILE_CODES  16
#define TILES       (NUM_CODES / TILE_CODES)   // 512
#define D_STEPS     (EMBED_DIM / 4)   // 64 WMMA k-steps per tile
#define LDS_STRIDE  260               // 256 + 4 pad -> bank-conflict-free B reads
#define WAVES_PER_BLOCK 4
#define BLOCK_THREADS   (WAVES_PER_BLOCK * 32)

// ---------------------------------------------------------------------------
// CDNA5 async global->LDS copy (ASYNCcnt) with inline-asm fallbacks.
// Builtin signature (per hipcc diagnostic): param0 = v4i* in device/global AS,
// param1 = LDS pointer, then imm offset, imm cpol.
// ---------------------------------------------------------------------------
#if defined(__has_builtin) && __has_builtin(__builtin_amdgcn_global_load_async_to_lds_b128)
__device__ __forceinline__ void async_copy_b128(const float* g, float* l) {
    __builtin_amdgcn_global_load_async_to_lds_b128(
        (__attribute__((address_space(1))) v4i*)(unsigned long long)(const void*)g,
        (__attribute__((address_space(3))) v4i*)(void*)l,
        0, 0);
}
#else
__device__ __forceinline__ void async_copy_b128(const float* g, float* l) {
    unsigned lofs = (unsigned)(unsigned long long)
        ((__attribute__((address_space(3))) void*)l);
    asm volatile("global_load_async_to_lds_b128 %0, %1, off"
                 :: "v"(lofs), "v"((unsigned long long)(const void*)g)
                 : "memory");
}
#endif

#if defined(__has_builtin) && __has_builtin(__builtin_amdgcn_s_wait_asynccnt)
#define WAIT_ASYNC0() __builtin_amdgcn_s_wait_asynccnt(0)
#else
#define WAIT_ASYNC0() asm volatile("s_wait_asynccnt 0x0" ::: "memory")
#endif

// ---------------------------------------------------------------------------
// Kernel 1: per-code squared norms ||e||^2 ; also zero the loss accumulator.
// ---------------------------------------------------------------------------
__global__ void vq_enorm_kernel(const float* __restrict__ codebook,
                                float* __restrict__ enorm,
                                float* __restrict__ loss_acc) {
    if (blockIdx.x == 0 && threadIdx.x == 0) *loss_acc = 0.0f;
    const int wave = threadIdx.x >> 5;
    const int lane = threadIdx.x & 31;
    const int code = blockIdx.x * (blockDim.x >> 5) + wave;
    if (code >= NUM_CODES) return;
    const float* row = codebook + (size_t)code * EMBED_DIM;
    float s = 0.0f;
    #pragma unroll
    for (int d = 0; d < EMBED_DIM; d += 32) {
        float v = row[d + lane];
        s += v * v;
    }
    #pragma unroll
    for (int m = 16; m >= 1; m >>= 1) s += __shfl_xor(s, m, 32);
    if (lane == 0) enorm[code] = s;
}

// ---------------------------------------------------------------------------
// Kernel 2: fused GEMM (x . codebook^T via V_WMMA_F32_16X16X4_F32) + argmin.
// Double-buffered LDS; codebook tiles streamed with async-to-LDS copies so
// tile t+2's fetch overlaps tile t+1's 64-WMMA compute phase.
// Argmin score = ||e||^2 - 2*dot  (||x||^2 is row-constant, dropped).
// ---------------------------------------------------------------------------
__device__ __forceinline__ void issue_tile_async(const float* __restrict__ codebook,
                                                 int cBase, float* dst, int tid) {
    // 16 codes x 256 floats = 1024 float4 slots; 8 async b128 per thread.
    #pragma unroll
    for (int k = 0; k < 8; ++k) {
        int f  = tid + k * BLOCK_THREADS;   // 0..1023
        int r  = f >> 6;                    // code row within tile
        int c4 = f & 63;                    // float4 column
        async_copy_b128(codebook + (size_t)(cBase + r) * EMBED_DIM + c4 * 4,
                        dst + r * LDS_STRIDE + c4 * 4);
    }
}

__global__ __launch_bounds__(BLOCK_THREADS)
void vq_argmin_kernel(const float* __restrict__ x,
                      const float* __restrict__ codebook,
                      const float* __restrict__ enorm,
                      int* __restrict__ idx_out) {
    __shared__ __align__(16) float lds[2][TILE_CODES * LDS_STRIDE];

    const int tid  = threadIdx.x;
    const int wave = tid >> 5;
    const int lane = tid & 31;
    const int half = lane >> 4;     // 0: K=0,1 / rows 0-7 ; 1: K=2,3 / rows 8-15
    const int m16  = lane & 15;     // A row index / B column index
    const int rowBase = (blockIdx.x * WAVES_PER_BLOCK + wave) * 16;

    // Kick off tiles 0 and 1 asynchronously before anything else.
    issue_tile_async(codebook, 0,          lds[0], tid);
    issue_tile_async(codebook, TILE_CODES, lds[1], tid);

    // --- A fragments (overlapped with the async copies): 16x4 f32 layout:
    //     lane<16: M=lane, K={0,1}; lane>=16: M=lane-16, K={2,3}
    v2f a[D_STEPS];
    const float* xrow = x + (size_t)(rowBase + m16) * EMBED_DIM + half * 2;
    #pragma unroll
    for (int s = 0; s < D_STEPS; ++s)
        a[s] = *(const v2f*)(xrow + s * 4);

    float best[8];
    int   bestIdx[8];
    #pragma unroll
    for (int j = 0; j < 8; ++j) { best[j] = 3.4e38f; bestIdx[j] = 0; }

    WAIT_ASYNC0();      // tiles 0 and 1 landed in LDS
    __syncthreads();

    for (int t = 0; t < TILES; ++t) {
        const int   cBase = t * TILE_CODES;
        const int   cur   = t & 1;
        const float* buf  = lds[cur];

        // --- 64 x V_WMMA_F32_16X16X4_F32 over D=256
        v8f acc = {};
        const float* bptr = &buf[m16 * LDS_STRIDE + half * 2];
        #pragma unroll
        for (int s = 0; s < D_STEPS; ++s) {
            v2f b = *(const v2f*)(bptr + s * 4);
            acc = __builtin_amdgcn_wmma_f32_16x16x4_f32(
                      false, a[s], false, b, (short)0, acc, false, false);
        }

        // C layout: VGPR j, lanes 0-15 -> row j, col lane; lanes 16-31 -> row j+8
        const float en = enorm[cBase + m16];
        #pragma unroll
        for (int j = 0; j < 8; ++j) {
            float score = en - 2.0f * acc[j];
            if (score < best[j]) { best[j] = score; bestIdx[j] = cBase + m16; }
        }

        // Pipeline: tile t+1's async writes (issued at t-1) complete; publish;
        // then refill this buffer with tile t+2.
        WAIT_ASYNC0();
        __syncthreads();
        if (t + 2 < TILES)
            issue_tile_async(codebook, (t + 2) * TILE_CODES, lds[cur], tid);
    }

    // Cross-lane argmin within each 16-lane half (same matrix row per VGPR j).
    #pragma unroll
    for (int j = 0; j < 8; ++j) {
        float b  = best[j];
        int   bi = bestIdx[j];
        #pragma unroll
        for (int m = 8; m >= 1; m >>= 1) {
            float ob  = __shfl_xor(b,  m, 32);
            int   obi = __shfl_xor(bi, m, 32);
            if (ob < b || (ob == b && obi < bi)) { b = ob; bi = obi; }
        }
        if (m16 == 0)
            idx_out[rowBase + j + half * 8] = bi;
    }
}

// ---------------------------------------------------------------------------
// Kernel 3: gather quantized rows (== straight-through output numerically),
// write indices (as float), accumulate commitment-loss sum of (q - x)^2.
// ---------------------------------------------------------------------------
__global__ __launch_bounds__(256)
void vq_gather_kernel(const float* __restrict__ x,
                      const float* __restrict__ codebook,
                      const int* __restrict__ idx_in,
                      float* __restrict__ q_out,
                      float* __restrict__ idx_f_out,
                      float* __restrict__ loss_acc) {
    const int row = blockIdx.x;
    const int d   = threadIdx.x;           // 256 threads = EMBED_DIM
    const int code = idx_in[row];

    float q  = codebook[(size_t)code * EMBED_DIM + d];
    float xv = x[(size_t)row * EMBED_DIM + d];
    q_out[(size_t)row * EMBED_DIM + d] = q;
    if (d == 0) idx_f_out[row] = (float)code;

    float diff = q - xv;
    float s = diff * diff;
    #pragma unroll
    for (int m = 16; m >= 1; m >>= 1) s += __shfl_xor(s, m, 32);

    __shared__ float partial[8];
    const int wv = d >> 5, lane = d & 31;
    if (lane == 0) partial[wv] = s;
    __syncthreads();
    if (d == 0) {
        float tot = 0.0f;
        #pragma unroll
        for (int w = 0; w < 8; ++w) tot += partial[w];
        atomicAdd(loss_acc, tot);
    }
}

// ---------------------------------------------------------------------------
// Kernel 4: finalize commitment loss = 0.25 * mean((q - x)^2)
// ---------------------------------------------------------------------------
__global__ void vq_loss_kernel(const float* __restrict__ loss_acc,
                               float* __restrict__ out_loss) {
    *out_loss = 0.25f * (*loss_acc) / 4194304.0f;   // N_ROWS * EMBED_DIM
}

// ---------------------------------------------------------------------------
extern "C" void kernel_launch(void* const* d_in, const int* in_sizes, int n_in,
                              void* d_out, int out_size, void* d_ws, size_t ws_size,
                              hipStream_t stream) {
    const float* x        = (const float*)d_in[0];   // [16,1024,256] f32
    const float* codebook = (const float*)d_in[1];   // [8192,256]   f32

    // workspace layout
    float* enorm    = (float*)d_ws;                                   // 8192 f32
    int*   idx      = (int*)((char*)d_ws + NUM_CODES * sizeof(float));// 16384 i32
    float* loss_acc = (float*)((char*)d_ws + NUM_CODES * sizeof(float)
                                           + N_ROWS * sizeof(int));   // 1 f32

    // output layout: quantized_st | indices | loss
    float* q_out    = (float*)d_out;
    float* idx_out  = q_out + (size_t)N_ROWS * EMBED_DIM;
    float* loss_out = idx_out + N_ROWS;

    vq_enorm_kernel<<<NUM_CODES / 8, 256, 0, stream>>>(codebook, enorm, loss_acc);
    vq_argmin_kernel<<<N_ROWS / (16 * WAVES_PER_BLOCK), BLOCK_THREADS, 0, stream>>>(
        x, codebook, enorm, idx);
    vq_gather_kernel<<<N_ROWS, 256, 0, stream>>>(x, codebook, idx,
                                                 q_out, idx_out, loss_acc);
    vq_loss_kernel<<<1, 1, 0, stream>>>(loss_acc, loss_out);
}